// GruEncoder_57844619543126
// MI455X (gfx1250) — compile-verified
//
#include <hip/hip_runtime.h>
#include <hip/hip_bf16.h>

#define Bb   256
#define Tt   128
#define NINn 1024
#define H1n  512
#define H2n  512
#define En   256
#define An   16

// GEMM tiling
#define BM   64
#define BN   128
#define KT   32
#define LPAD 34   // LDS row stride (floats): even (keeps 8B alignment for v2f), padded vs 32

typedef float v2f __attribute__((ext_vector_type(2)));
typedef float v8f __attribute__((ext_vector_type(8)));

__device__ __forceinline__ v8f wmma_f32(v2f a, v2f b, v8f c) {
  // V_WMMA_F32_16X16X4_F32: (neg_a, A, neg_b, B, c_mod, C, reuse_a, reuse_b)
  return __builtin_amdgcn_wmma_f32_16x16x4_f32(false, a, false, b, (short)0, c, false, false);
}

// D = A(MxK) @ B(KxN) + bias, row-major fp32.
// Block 256 thr (8 waves) -> 64(M) x 128(N) tile; wave tile 16x64 (4 C tiles).
// K staged through LDS in KT=32 slices, double buffered:
//   As[buf][64][KT]  (row-major, stride LPAD)
//   Bs[buf][128][KT] (K-transposed so a B fragment {B[k][c],B[k+1][c]} is one ds_load_b64)
// Requires: M%64==0, N%128==0, K%32==0 (true for all call sites).
__global__ __launch_bounds__(256)
void gemm_wmma_f32(const float* __restrict__ A, const float* __restrict__ B,
                   const float* __restrict__ bias, float* __restrict__ C,
                   int M, int N, int K) {
  __shared__ __align__(16) float As[2][BM * LPAD];
  __shared__ __align__(16) float Bs[2][BN * LPAD];

  const int tid   = threadIdx.x;
  const int wid   = tid >> 5;
  const int lane  = tid & 31;
  const int r16   = lane & 15;
  const int khalf = lane >> 4;               // 0 or 1
  const int waveM = wid >> 1;                // 0..3
  const int waveN = wid & 1;                 // 0..1

  const long rowBlk = (long)blockIdx.y * BM;
  const int  colBlk = blockIdx.x * BN;

  // staging geometry (per thread): A = 2 float4, B = 4 float4
  const int arow0 = tid >> 3;                // l>>3, l=tid
  const int ak4   = (tid & 7) << 2;
  const int bk0   = tid >> 5;                // l>>5, l=tid (+8 per i)
  const int bc4   = (tid & 31) << 2;

  float4 ra[2], rb[4];

  // ---- prologue: load tile kt=0 and stage into buffer 0 ----
#pragma unroll
  for (int i = 0; i < 2; i++)
    ra[i] = *(const float4*)(A + (rowBlk + arow0 + i * 32) * (long)K + ak4);
#pragma unroll
  for (int i = 0; i < 4; i++)
    rb[i] = *(const float4*)(B + (long)(bk0 + i * 8) * N + colBlk + bc4);
#pragma unroll
  for (int i = 0; i < 2; i++) {
    float* d = &As[0][(arow0 + i * 32) * LPAD + ak4];
    d[0] = ra[i].x; d[1] = ra[i].y; d[2] = ra[i].z; d[3] = ra[i].w;
  }
#pragma unroll
  for (int i = 0; i < 4; i++) {
    const int kr = bk0 + i * 8;
    Bs[0][(bc4 + 0) * LPAD + kr] = rb[i].x;
    Bs[0][(bc4 + 1) * LPAD + kr] = rb[i].y;
    Bs[0][(bc4 + 2) * LPAD + kr] = rb[i].z;
    Bs[0][(bc4 + 3) * LPAD + kr] = rb[i].w;
  }

  v8f acc0 = {0,0,0,0,0,0,0,0}, acc1 = acc0, acc2 = acc0, acc3 = acc0;

  int cur = 0;
  for (int kt = 0; kt < K; kt += KT) {
    __syncthreads();                         // staging of `cur` visible to all waves

    const bool more = (kt + KT) < K;
    if (more) {                              // overlap next tile's global loads with WMMA
      const int kn = kt + KT;
#pragma unroll
      for (int i = 0; i < 2; i++)
        ra[i] = *(const float4*)(A + (rowBlk + arow0 + i * 32) * (long)K + kn + ak4);
#pragma unroll
      for (int i = 0; i < 4; i++)
        rb[i] = *(const float4*)(B + (long)(kn + bk0 + i * 8) * N + colBlk + bc4);
    }

    const float* __restrict__ ar  = &As[cur][(waveM * 16 + r16) * LPAD];
    const float* __restrict__ br0 = &Bs[cur][(waveN * 64 + r16     ) * LPAD];
    const float* __restrict__ br1 = &Bs[cur][(waveN * 64 + r16 + 16) * LPAD];
    const float* __restrict__ br2 = &Bs[cur][(waveN * 64 + r16 + 32) * LPAD];
    const float* __restrict__ br3 = &Bs[cur][(waveN * 64 + r16 + 48) * LPAD];
#pragma unroll
    for (int kk = 0; kk < KT; kk += 4) {
      const int ka = kk + 2 * khalf;
      v2f a  = *(const v2f*)(ar  + ka);
      v2f b0 = *(const v2f*)(br0 + ka);
      v2f b1 = *(const v2f*)(br1 + ka);
      v2f b2 = *(const v2f*)(br2 + ka);
      v2f b3 = *(const v2f*)(br3 + ka);
      acc0 = wmma_f32(a, b0, acc0);
      acc1 = wmma_f32(a, b1, acc1);
      acc2 = wmma_f32(a, b2, acc2);
      acc3 = wmma_f32(a, b3, acc3);
    }

    if (more) {                              // stage next tile into the other buffer
      const int nxt = cur ^ 1;
#pragma unroll
      for (int i = 0; i < 2; i++) {
        float* d = &As[nxt][(arow0 + i * 32) * LPAD + ak4];
        d[0] = ra[i].x; d[1] = ra[i].y; d[2] = ra[i].z; d[3] = ra[i].w;
      }
#pragma unroll
      for (int i = 0; i < 4; i++) {
        const int kr = bk0 + i * 8;
        Bs[nxt][(bc4 + 0) * LPAD + kr] = rb[i].x;
        Bs[nxt][(bc4 + 1) * LPAD + kr] = rb[i].y;
        Bs[nxt][(bc4 + 2) * LPAD + kr] = rb[i].z;
        Bs[nxt][(bc4 + 3) * LPAD + kr] = rb[i].w;
      }
      cur = nxt;
    }
  }

  const long m0 = rowBlk + waveM * 16 + khalf * 8;
  const int  c0 = colBlk + waveN * 64 + r16;
  const float bv0 = bias[c0], bv1 = bias[c0 + 16], bv2 = bias[c0 + 32], bv3 = bias[c0 + 48];
#pragma unroll
  for (int v = 0; v < 8; v++) {
    float* crow = C + (m0 + v) * (long)N;
    crow[c0     ] = acc0[v] + bv0;
    crow[c0 + 16] = acc1[v] + bv1;
    crow[c0 + 32] = acc2[v] + bv2;
    crow[c0 + 48] = acc3[v] + bv3;
  }
}

// henc = concat(H[256x256], AEprev[256x16]) @ Whe[272x256] + bhe  -> HENC[256x256]
// Small latency-bound serial GEMM: direct global fragment loads (no barriers).
// AEprev row for batch b lives at AEprev + b*(T*16); null at t==0 (a_prev == 0).
__global__ __launch_bounds__(256)
void gemm_henc(const float* __restrict__ H, const float* __restrict__ AEprev,
               const float* __restrict__ Whe, const float* __restrict__ bhe,
               float* __restrict__ HENC) {
  const int tid   = threadIdx.x;
  const int wid   = tid >> 5;
  const int lane  = tid & 31;
  const int r16   = lane & 15;
  const int khalf = lane >> 4;
  const int waveM = wid >> 1;
  const int waveN = wid & 1;

  const int rowBase = blockIdx.y * 64 + waveM * 16;
  const int colBase = blockIdx.x * 128 + waveN * 64;
  const int c0 = colBase + r16;
  const int N = En;

  v8f acc0 = {0,0,0,0,0,0,0,0}, acc1 = acc0, acc2 = acc0, acc3 = acc0;

  const float* __restrict__ Arow = H + (rowBase + r16) * (long)En;
  for (int k0 = 0; k0 < En; k0 += 4) {
    const int ka = k0 + 2 * khalf;
    v2f a = *(const v2f*)(Arow + ka);
    const float* __restrict__ Br0 = Whe + (long)ka * N;
    const float* __restrict__ Br1 = Br0 + N;
    v2f b0 = { Br0[c0     ], Br1[c0     ] };
    v2f b1 = { Br0[c0 + 16], Br1[c0 + 16] };
    v2f b2 = { Br0[c0 + 32], Br1[c0 + 32] };
    v2f b3 = { Br0[c0 + 48], Br1[c0 + 48] };
    acc0 = wmma_f32(a, b0, acc0);
    acc1 = wmma_f32(a, b1, acc1);
    acc2 = wmma_f32(a, b2, acc2);
    acc3 = wmma_f32(a, b3, acc3);
  }
  if (AEprev != nullptr) {
    const float* __restrict__ A2row = AEprev + (rowBase + r16) * (long)(Tt * An);
    const float* __restrict__ Wact  = Whe + (long)En * N;   // rows 256..271
    for (int k0 = 0; k0 < An; k0 += 4) {
      const int ka = k0 + 2 * khalf;
      v2f a = *(const v2f*)(A2row + ka);
      const float* __restrict__ Br0 = Wact + (long)ka * N;
      const float* __restrict__ Br1 = Br0 + N;
      v2f b0 = { Br0[c0     ], Br1[c0     ] };
      v2f b1 = { Br0[c0 + 16], Br1[c0 + 16] };
      v2f b2 = { Br0[c0 + 32], Br1[c0 + 32] };
      v2f b3 = { Br0[c0 + 48], Br1[c0 + 48] };
      acc0 = wmma_f32(a, b0, acc0);
      acc1 = wmma_f32(a, b1, acc1);
      acc2 = wmma_f32(a, b2, acc2);
      acc3 = wmma_f32(a, b3, acc3);
    }
  }

  const int m0 = rowBase + khalf * 8;
  const float bv0 = bhe[c0], bv1 = bhe[c0 + 16], bv2 = bhe[c0 + 32], bv3 = bhe[c0 + 48];
#pragma unroll
  for (int v = 0; v < 8; v++) {
    float* crow = HENC + (m0 + v) * (long)N;
    crow[c0     ] = acc0[v] + bv0;
    crow[c0 + 16] = acc1[v] + bv1;
    crow[c0 + 32] = acc2[v] + bv2;
    crow[c0 + 48] = acc3[v] + bv3;
  }
}

// BatchNorm (training mode, biased var, per-timestep batch stats) + ELU, in place.
// Y layout [B*T, N], row = b*T + t. Block: 8 cols x 32 lanes; lane owns 8 batch rows.
__global__ __launch_bounds__(256)
void bn_elu_kernel(float* __restrict__ Y, const float* __restrict__ gamma,
                   const float* __restrict__ beta, int N) {
  const int t    = blockIdx.x;
  const int col  = blockIdx.y * 8 + (threadIdx.x >> 5);
  const int lane = threadIdx.x & 31;

  float vals[8];
  float s = 0.f, s2 = 0.f;
#pragma unroll
  for (int i = 0; i < 8; i++) {
    const int b = lane + i * 32;
    float v = Y[((long)b * Tt + t) * N + col];
    vals[i] = v;
    s += v; s2 += v * v;
  }
#pragma unroll
  for (int off = 16; off >= 1; off >>= 1) {
    s  += __shfl_xor(s,  off, 32);
    s2 += __shfl_xor(s2, off, 32);
  }
  const float mean = s * (1.f / 256.f);
  const float var  = s2 * (1.f / 256.f) - mean * mean;
  const float inv  = rsqrtf(var + 1e-5f);
  const float g = gamma[col] * inv, be = beta[col];
#pragma unroll
  for (int i = 0; i < 8; i++) {
    const int b = lane + i * 32;
    float x = g * (vals[i] - mean) + be;
    Y[((long)b * Tt + t) * N + col] = (x > 0.f) ? x : expm1f(x);
  }
}

// AE = actions @ Wae + bae   ([B*T,16] x [16,16])
__global__ __launch_bounds__(256)
void ae_kernel(const float* __restrict__ act, const float* __restrict__ Wae,
               const float* __restrict__ bae, float* __restrict__ AE, int total) {
  const int idx = blockIdx.x * blockDim.x + threadIdx.x;
  if (idx >= total) return;
  const int row = idx >> 4, j = idx & 15;
  const float* a = act + (long)row * 16;
  float s = bae[j];
#pragma unroll
  for (int k = 0; k < 16; k++) s += a[k] * Wae[k * 16 + j];
  AE[idx] = s;
}

// GRU gates: h = (1-z)*n + z*henc; gi precomputed (incl. bih), gh (incl. bhh).
__global__ __launch_bounds__(256)
void gru_gates(const float* __restrict__ GI, const float* __restrict__ GH,
               const float* __restrict__ HENC, float* __restrict__ Hout, int t) {
  const int idx = blockIdx.x * blockDim.x + threadIdx.x;   // 0..65535
  const int b = idx >> 8, j = idx & 255;
  const long gir = ((long)b * Tt + t) * (3 * En);
  const long ghr = (long)b * (3 * En);
  const float ir = GI[gir + j],  iz = GI[gir + En + j],  inn = GI[gir + 2 * En + j];
  const float hr = GH[ghr + j],  hz = GH[ghr + En + j],  hn  = GH[ghr + 2 * En + j];
  const float r = 1.f / (1.f + __expf(-(ir + hr)));
  const float z = 1.f / (1.f + __expf(-(iz + hz)));
  const float n = tanhf(inn + r * hn);
  Hout[idx] = (1.f - z) * n + z * HENC[idx];
}

extern "C" void kernel_launch(void* const* d_in, const int* in_sizes, int n_in,
                              void* d_out, int out_size, void* d_ws, size_t ws_size,
                              hipStream_t stream) {
  const float* obs  = (const float*)d_in[0];
  const float* acts = (const float*)d_in[1];
  const float* W1   = (const float*)d_in[2];
  const float* b1   = (const float*)d_in[3];
  const float* g1   = (const float*)d_in[4];
  const float* be1  = (const float*)d_in[5];
  const float* W2   = (const float*)d_in[6];
  const float* b2   = (const float*)d_in[7];
  const float* g2   = (const float*)d_in[8];
  const float* be2  = (const float*)d_in[9];
  const float* W3   = (const float*)d_in[10];
  const float* b3   = (const float*)d_in[11];
  const float* Wih  = (const float*)d_in[12];
  const float* bih  = (const float*)d_in[13];
  const float* Whh  = (const float*)d_in[14];
  const float* bhh  = (const float*)d_in[15];
  const float* Whe  = (const float*)d_in[16];
  const float* bhe  = (const float*)d_in[17];
  const float* Wae  = (const float*)d_in[18];
  const float* bae  = (const float*)d_in[19];
  float* out = (float*)d_out;

  const long ROWS = (long)Bb * Tt;            // 32768 (row index = b*T + t)
  float* ws = (float*)d_ws;
  float* Y1   = ws;                 ws += ROWS * H1n;
  float* Y2   = ws;                 ws += ROWS * H2n;
  float* X    = ws;                 ws += ROWS * En;
  float* GI   = ws;                 ws += ROWS * 3 * En;
  float* AE   = ws;                 ws += ROWS * An;
  float* HENC = ws;                 ws += (long)Bb * En;
  float* GH   = ws;                 ws += (long)Bb * 3 * En;
  float* H    = ws;                 ws += (long)Bb * En;
  (void)ws_size; (void)n_in; (void)in_sizes; (void)out_size;

  hipMemsetAsync(H, 0, (size_t)Bb * En * sizeof(float), stream);

  dim3 blk(256);
  // ---- time-parallel phase (81% of FLOPs, fully batched over T*B rows) ----
  gemm_wmma_f32<<<dim3(H1n / BN, ROWS / BM), blk, 0, stream>>>(obs, W1, b1, Y1, ROWS, H1n, NINn);
  bn_elu_kernel <<<dim3(Tt, H1n / 8), blk, 0, stream>>>(Y1, g1, be1, H1n);
  gemm_wmma_f32<<<dim3(H2n / BN, ROWS / BM), blk, 0, stream>>>(Y1, W2, b2, Y2, ROWS, H2n, H1n);
  bn_elu_kernel <<<dim3(Tt, H2n / 8), blk, 0, stream>>>(Y2, g2, be2, H2n);
  gemm_wmma_f32<<<dim3(En / BN, ROWS / BM), blk, 0, stream>>>(Y2, W3, b3, X, ROWS, En, H2n);
  gemm_wmma_f32<<<dim3(3 * En / BN, ROWS / BM), blk, 0, stream>>>(X, Wih, bih, GI, ROWS, 3 * En, En);
  ae_kernel     <<<dim3((int)((ROWS * An + 255) / 256)), blk, 0, stream>>>(acts, Wae, bae, AE, (int)(ROWS * An));

  // ---- sequential recurrence (latency-bound; 3 small launches per step) ----
  for (int t = 0; t < Tt; t++) {
    const float* aeprev = (t == 0) ? nullptr : (AE + (long)(t - 1) * An);
    gemm_henc    <<<dim3(En / BN, Bb / BM), blk, 0, stream>>>(H, aeprev, Whe, bhe, HENC);
    gemm_wmma_f32<<<dim3(3 * En / BN, Bb / BM), blk, 0, stream>>>(HENC, Whh, bhh, GH, Bb, 3 * En, En);
    float* hout = (t == Tt - 1) ? out : H;
    gru_gates    <<<dim3(Bb * En / 256), blk, 0, stream>>>(GI, GH, HENC, hout, t);
  }
}